// PointNetFeaturePropagation_39316130627752
// MI455X (gfx1250) — compile-verified
//
#include <hip/hip_runtime.h>
#include <hip/hip_bf16.h>

typedef float v2f __attribute__((ext_vector_type(2)));
typedef float v8f __attribute__((ext_vector_type(8)));
typedef int   v4i __attribute__((ext_vector_type(4)));

#define B_   8
#define N_   8192
#define S_   2048
#define D1_  128
#define D2_  256
#define K1_  384          // D1 + D2
#define C1_  256          // layer-1 out channels
#define C2_  128          // layer-2 out channels
#define M_   (B_ * N_)    // 65536 rows

#define BN_EPS     1e-5f
#define INTERP_EPS 1e-8f

// GEMM tiling
#define TM_   128         // rows per block
#define TN_   64          // cols per block
#define KC_   32          // K-chunk staged in LDS
#define LDSR_ 36          // padded LDS row stride (floats): 16B-aligned, bank-safe

// Prefer the CDNA5 async global->LDS engine (ASYNCcnt path, no VGPR round
// trip) when the toolchain exposes it; otherwise fall back to reg staging.
#if defined(__gfx1250__) &&                                                   \
    __has_builtin(__builtin_amdgcn_global_load_async_to_lds_b128) &&          \
    __has_builtin(__builtin_amdgcn_s_wait_asynccnt)
#define USE_ASYNC_LDS 1
#else
#define USE_ASYNC_LDS 0
#endif

// ---------------------------------------------------------------------------
// 0. zero the BN sum/sumsq accumulators (must happen every call: graph replay)
// ---------------------------------------------------------------------------
__global__ void zero_stats_kernel(float* stats) {
    stats[blockIdx.x * blockDim.x + threadIdx.x] = 0.0f;
}

// ---------------------------------------------------------------------------
// 1. 3-NN search: xyz2 for one batch staged in LDS (24 KB), one thread/query
// ---------------------------------------------------------------------------
__global__ void knn3_kernel(const float* __restrict__ xyz1,
                            const float* __restrict__ xyz2,
                            int* __restrict__ idx, float* __restrict__ wgt) {
    __shared__ float sx[S_], sy[S_], sz[S_];
    const int b = blockIdx.y;
    const float* x2 = xyz2 + (long)b * 3 * S_;
    for (int s = threadIdx.x; s < S_; s += blockDim.x) {
        sx[s] = x2[s];
        sy[s] = x2[S_ + s];
        sz[s] = x2[2 * S_ + s];
    }
    __syncthreads();

    const int n = blockIdx.x * blockDim.x + threadIdx.x;
    const float* x1 = xyz1 + (long)b * 3 * N_;
    const float px = x1[n], py = x1[N_ + n], pz = x1[2 * N_ + n];

    float d0 = 3.4e38f, d1 = 3.4e38f, d2 = 3.4e38f;
    int   i0 = 0, i1 = 0, i2 = 0;
    for (int s = 0; s < S_; ++s) {
        const float dx = px - sx[s], dy = py - sy[s], dz = pz - sz[s];
        const float d = dx * dx + dy * dy + dz * dz;
        if (d < d0)      { d2 = d1; i2 = i1; d1 = d0; i1 = i0; d0 = d; i0 = s; }
        else if (d < d1) { d2 = d1; i2 = i1; d1 = d;  i1 = s; }
        else if (d < d2) { d2 = d;  i2 = s; }
    }
    const float r0 = 1.0f / (d0 + INTERP_EPS);
    const float r1 = 1.0f / (d1 + INTERP_EPS);
    const float r2 = 1.0f / (d2 + INTERP_EPS);
    const float inv = 1.0f / (r0 + r1 + r2);

    const long base = ((long)b * N_ + n) * 3;
    idx[base + 0] = i0;  idx[base + 1] = i1;  idx[base + 2] = i2;
    wgt[base + 0] = r0 * inv; wgt[base + 1] = r1 * inv; wgt[base + 2] = r2 * inv;
}

// ---------------------------------------------------------------------------
// 2. LDS-tiled transpose: src [B][C][L]  ->  dst[((b*L)+l)*ldd + off + c]
//    (C and L are multiples of 32). block = (32,8)
// ---------------------------------------------------------------------------
__global__ void transpose_kernel(const float* __restrict__ src,
                                 float* __restrict__ dst,
                                 int C, int L, int ldd, int off) {
    __shared__ float tile[32][33];
    const int b  = blockIdx.z;
    const int c0 = blockIdx.y * 32;
    const int l0 = blockIdx.x * 32;
    for (int i = threadIdx.y; i < 32; i += 8)
        tile[i][threadIdx.x] = src[((long)b * C + c0 + i) * L + l0 + threadIdx.x];
    __syncthreads();
    for (int i = threadIdx.y; i < 32; i += 8)
        dst[((long)b * L + l0 + i) * (long)ldd + off + c0 + threadIdx.x] =
            tile[threadIdx.x][i];
}

// ---------------------------------------------------------------------------
// 3. interp: H[m][128 + c] = sum_k w_k * P2T[b][idx_k][c]  (block = 256 = D2)
// ---------------------------------------------------------------------------
__global__ void interp_kernel(const float* __restrict__ P2T,
                              const int* __restrict__ idx,
                              const float* __restrict__ wgt,
                              float* __restrict__ H) {
    const long m = blockIdx.x;
    const int  b = (int)(m / N_);
    const int  c = threadIdx.x;
    const long base = m * 3;
    const int  i0 = idx[base], i1 = idx[base + 1], i2 = idx[base + 2];
    const float w0 = wgt[base], w1 = wgt[base + 1], w2 = wgt[base + 2];
    const float* p = P2T + (long)b * S_ * D2_;
    const float v = w0 * p[(long)i0 * D2_ + c]
                  + w1 * p[(long)i1 * D2_ + c]
                  + w2 * p[(long)i2 * D2_ + c];
    H[m * K1_ + D1_ + c] = v;
}

// ---------------------------------------------------------------------------
// 4. WMMA f32 GEMM with LDS double buffering:
//      Y[M x Nout] = A[M x K] * W^T + bias,  W is [Nout x K] row-major.
//    block = 256 threads (8 waves); each wave -> 16 rows x 64 cols (4 tiles)
//    grid  = (Nout/64, M/128); K % 32 == 0.
//
//    Per K-chunk (32): A(128x32) and W(64x32) tiles live in LDS (padded row
//    stride 36 floats: b128 aligned, ds_load_b64 aligned, bank-conflict-free
//    fragment reads). The next chunk is fetched with the CDNA5 async
//    global->LDS engine (ASYNCcnt) while 32 WMMAs run on the current buffer;
//    s_wait_asynccnt 0 + one barrier per chunk.
// ---------------------------------------------------------------------------
__global__ void gemm_wmma_kernel(const float* __restrict__ A,
                                 const float* __restrict__ W,
                                 const float* __restrict__ bias,
                                 float* __restrict__ Y,
                                 int K, int Nout) {
    __shared__ float As[2][TM_ * LDSR_];
    __shared__ float Bs[2][TN_ * LDSR_];

    const int tid  = threadIdx.x;
    const int wave = tid >> 5;
    const int lane = tid & 31;
    const int hlf  = lane >> 4;     // which K-pair this lane holds
    const int lrow = lane & 15;

    const long mblk = (long)blockIdx.y * TM_;
    const int  nblk = blockIdx.x * TN_;

    const float* Ablk = A + mblk * (long)K;
    const float* Wblk = W + (long)nblk * K;

#if USE_ASYNC_LDS
    typedef __attribute__((address_space(1))) v4i* gvec_t;
    typedef __attribute__((address_space(3))) v4i* lvec_t;
    // issue 6 async b128 global->LDS copies per thread (4 A-tile + 2 W-tile)
    auto issue_chunk = [&](int buf, int k0) {
#pragma unroll
        for (int i = 0; i < 4; ++i) {
            const int f = i * 256 + tid;          // [0,1024)
            const int row = f >> 3, c4 = (f & 7) << 2;
            __builtin_amdgcn_global_load_async_to_lds_b128(
                (gvec_t)(Ablk + (long)row * K + k0 + c4),
                (lvec_t)(&As[buf][row * LDSR_ + c4]), 0, 0);
        }
#pragma unroll
        for (int i = 0; i < 2; ++i) {
            const int f = i * 256 + tid;          // [0,512)
            const int row = f >> 3, c4 = (f & 7) << 2;
            __builtin_amdgcn_global_load_async_to_lds_b128(
                (gvec_t)(Wblk + (long)row * K + k0 + c4),
                (lvec_t)(&Bs[buf][row * LDSR_ + c4]), 0, 0);
        }
    };
#else
    float4 ra[4], rw[2];
    auto load_chunk = [&](int k0) {
#pragma unroll
        for (int i = 0; i < 4; ++i) {
            const int f = i * 256 + tid;
            const int row = f >> 3, c4 = (f & 7) << 2;
            ra[i] = *(const float4*)(Ablk + (long)row * K + k0 + c4);
        }
#pragma unroll
        for (int i = 0; i < 2; ++i) {
            const int f = i * 256 + tid;
            const int row = f >> 3, c4 = (f & 7) << 2;
            rw[i] = *(const float4*)(Wblk + (long)row * K + k0 + c4);
        }
    };
    auto store_chunk = [&](int buf) {
#pragma unroll
        for (int i = 0; i < 4; ++i) {
            const int f = i * 256 + tid;
            const int row = f >> 3, c4 = (f & 7) << 2;
            *(float4*)(&As[buf][row * LDSR_ + c4]) = ra[i];
        }
#pragma unroll
        for (int i = 0; i < 2; ++i) {
            const int f = i * 256 + tid;
            const int row = f >> 3, c4 = (f & 7) << 2;
            *(float4*)(&Bs[buf][row * LDSR_ + c4]) = rw[i];
        }
    };
#endif

    v8f acc0 = {}, acc1 = {}, acc2 = {}, acc3 = {};

    const int nchunks = K / KC_;

#if USE_ASYNC_LDS
    issue_chunk(0, 0);
    __builtin_amdgcn_s_wait_asynccnt(0);
#else
    load_chunk(0);
    store_chunk(0);
#endif
    __syncthreads();

    const int arow = (wave * 16 + lrow) * LDSR_ + 2 * hlf;
    const int b0r  = ( 0 + lrow) * LDSR_ + 2 * hlf;
    const int b1r  = (16 + lrow) * LDSR_ + 2 * hlf;
    const int b2r  = (32 + lrow) * LDSR_ + 2 * hlf;
    const int b3r  = (48 + lrow) * LDSR_ + 2 * hlf;

    for (int c = 0; c < nchunks; ++c) {
        const int p = c & 1;
#if USE_ASYNC_LDS
        if (c + 1 < nchunks) issue_chunk((c + 1) & 1, (c + 1) * KC_);
#else
        if (c + 1 < nchunks) load_chunk((c + 1) * KC_);
#endif

        const float* __restrict__ ap = &As[p][arow];
        const float* __restrict__ bp = &Bs[p][0];
#pragma unroll
        for (int k = 0; k < KC_; k += 4) {
            const v2f a  = *(const v2f*)(ap + k);
            const v2f b0 = *(const v2f*)(bp + b0r + k);
            const v2f b1 = *(const v2f*)(bp + b1r + k);
            const v2f b2 = *(const v2f*)(bp + b2r + k);
            const v2f b3 = *(const v2f*)(bp + b3r + k);
            acc0 = __builtin_amdgcn_wmma_f32_16x16x4_f32(false, a, false, b0,
                                                         (short)0, acc0, false, false);
            acc1 = __builtin_amdgcn_wmma_f32_16x16x4_f32(false, a, false, b1,
                                                         (short)0, acc1, false, false);
            acc2 = __builtin_amdgcn_wmma_f32_16x16x4_f32(false, a, false, b2,
                                                         (short)0, acc2, false, false);
            acc3 = __builtin_amdgcn_wmma_f32_16x16x4_f32(false, a, false, b3,
                                                         (short)0, acc3, false, false);
        }
#if USE_ASYNC_LDS
        if (c + 1 < nchunks) __builtin_amdgcn_s_wait_asynccnt(0);
#else
        if (c + 1 < nchunks) store_chunk((c + 1) & 1);
#endif
        __syncthreads();
    }

    const float bi0 = bias[nblk +  0 + lrow];
    const float bi1 = bias[nblk + 16 + lrow];
    const float bi2 = bias[nblk + 32 + lrow];
    const float bi3 = bias[nblk + 48 + lrow];

#pragma unroll
    for (int vi = 0; vi < 8; ++vi) {
        const long row = mblk + wave * 16 + vi + 8 * hlf;
        float* yr = Y + row * (long)Nout;
        yr[nblk +  0 + lrow] = acc0[vi] + bi0;
        yr[nblk + 16 + lrow] = acc1[vi] + bi1;
        yr[nblk + 32 + lrow] = acc2[vi] + bi2;
        yr[nblk + 48 + lrow] = acc3[vi] + bi3;
    }
}

// ---------------------------------------------------------------------------
// 5. per-channel sum / sumsq partials (block = Nout threads, coalesced cols)
// ---------------------------------------------------------------------------
__global__ void colstats_kernel(const float* __restrict__ Y,
                                float* __restrict__ sum, float* __restrict__ sumsq,
                                int Nout, int mchunk) {
    const int  c  = threadIdx.x;
    const long m0 = (long)blockIdx.x * mchunk;
    float s = 0.0f, sq = 0.0f;
    for (long m = m0; m < m0 + mchunk; ++m) {
        const float v = Y[m * Nout + c];
        s += v; sq += v * v;
    }
    atomicAdd(&sum[c], s);
    atomicAdd(&sumsq[c], sq);
}

__global__ void finalize_stats_kernel(const float* __restrict__ sum,
                                      const float* __restrict__ sumsq,
                                      float* __restrict__ mean,
                                      float* __restrict__ invstd, float invcount) {
    const int c = threadIdx.x;
    const float mu  = sum[c] * invcount;
    const float var = sumsq[c] * invcount - mu * mu;
    mean[c]   = mu;
    invstd[c] = rsqrtf(var + BN_EPS);
}

// ---------------------------------------------------------------------------
// 6. BN + ReLU in place (block = Nout threads, grid = M rows)
// ---------------------------------------------------------------------------
__global__ void bnrelu_kernel(float* __restrict__ Y,
                              const float* __restrict__ mean,
                              const float* __restrict__ invstd,
                              const float* __restrict__ gamma,
                              const float* __restrict__ beta, int Nout) {
    const int  c = threadIdx.x;
    const long i = (long)blockIdx.x * Nout + c;
    const float v = gamma[c] * (Y[i] - mean[c]) * invstd[c] + beta[c];
    Y[i] = v > 0.0f ? v : 0.0f;
}

// ---------------------------------------------------------------------------
// 7. fused BN + ReLU + transpose-store:  Y2 [M x 128] -> out [B][128][N]
//    block = (32,8), grid = (N/32, C2/32, B)
// ---------------------------------------------------------------------------
__global__ void bnrelu_transpose_out_kernel(const float* __restrict__ Y2,
                                            const float* __restrict__ mean,
                                            const float* __restrict__ invstd,
                                            const float* __restrict__ gamma,
                                            const float* __restrict__ beta,
                                            float* __restrict__ out) {
    __shared__ float tile[32][33];
    const int b  = blockIdx.z;
    const int c0 = blockIdx.y * 32;
    const int n0 = blockIdx.x * 32;
    for (int i = threadIdx.y; i < 32; i += 8) {
        const int n = n0 + i;
        const int c = c0 + threadIdx.x;
        float v = Y2[((long)b * N_ + n) * C2_ + c];
        v = gamma[c] * (v - mean[c]) * invstd[c] + beta[c];
        tile[i][threadIdx.x] = v > 0.0f ? v : 0.0f;
    }
    __syncthreads();
    for (int i = threadIdx.y; i < 32; i += 8) {
        const int c = c0 + i;
        const int n = n0 + threadIdx.x;
        out[((long)b * C2_ + c) * N_ + n] = tile[threadIdx.x][i];
    }
}

// ---------------------------------------------------------------------------
// launch
// ---------------------------------------------------------------------------
extern "C" void kernel_launch(void* const* d_in, const int* in_sizes, int n_in,
                              void* d_out, int out_size, void* d_ws, size_t ws_size,
                              hipStream_t stream) {
    const float* xyz1    = (const float*)d_in[0];
    const float* xyz2    = (const float*)d_in[1];
    const float* points1 = (const float*)d_in[2];
    const float* points2 = (const float*)d_in[3];
    const float* W1      = (const float*)d_in[4];
    const float* b1      = (const float*)d_in[5];
    const float* gamma1  = (const float*)d_in[6];
    const float* beta1   = (const float*)d_in[7];
    const float* W2      = (const float*)d_in[8];
    const float* b2      = (const float*)d_in[9];
    const float* gamma2  = (const float*)d_in[10];
    const float* beta2   = (const float*)d_in[11];
    float* out = (float*)d_out;

    // ---- workspace layout (floats) -------------------------------------
    float* ws = (float*)d_ws;
    int*   idx   = (int*)ws;                       // 3*M ints
    float* wgt   = ws + 3L * M_;                   // 3*M floats
    float* H     = wgt + 3L * M_;                  // M*384
    float* P2T   = H + (long)M_ * K1_;             // B*S*256
    float* Y1    = P2T + (long)B_ * S_ * D2_;      // M*256
    float* stats = Y1 + (long)M_ * C1_;            // 2048 floats
    float* Y2    = H;                              // reuse H region: M*128

    float* s1_sum = stats;            // 256
    float* s1_sq  = stats + 256;      // 256
    float* s1_mu  = stats + 512;      // 256
    float* s1_is  = stats + 768;      // 256
    float* s2_sum = stats + 1024;     // 128
    float* s2_sq  = stats + 1152;     // 128
    float* s2_mu  = stats + 1280;     // 128
    float* s2_is  = stats + 1408;     // 128

    // 0. zero accumulators (graph-replay safe: done every call)
    zero_stats_kernel<<<8, 256, 0, stream>>>(stats);

    // 1. 3-NN
    knn3_kernel<<<dim3(N_ / 256, B_), 256, 0, stream>>>(xyz1, xyz2, idx, wgt);

    // 2. transposes: points1 -> H[:, 0:128], points2 -> P2T
    transpose_kernel<<<dim3(N_ / 32, D1_ / 32, B_), dim3(32, 8), 0, stream>>>(
        points1, H, D1_, N_, K1_, 0);
    transpose_kernel<<<dim3(S_ / 32, D2_ / 32, B_), dim3(32, 8), 0, stream>>>(
        points2, P2T, D2_, S_, D2_, 0);

    // 3. interpolation -> H[:, 128:384]
    interp_kernel<<<M_, D2_, 0, stream>>>(P2T, idx, wgt, H);

    // 4. GEMM1: Y1 = H * W1^T + b1   (M x 384 x 256)
    gemm_wmma_kernel<<<dim3(C1_ / TN_, M_ / TM_), 256, 0, stream>>>(
        H, W1, b1, Y1, K1_, C1_);

    // 5. BN1 stats + normalize + relu
    colstats_kernel<<<64, C1_, 0, stream>>>(Y1, s1_sum, s1_sq, C1_, M_ / 64);
    finalize_stats_kernel<<<1, C1_, 0, stream>>>(s1_sum, s1_sq, s1_mu, s1_is,
                                                 1.0f / (float)M_);
    bnrelu_kernel<<<M_, C1_, 0, stream>>>(Y1, s1_mu, s1_is, gamma1, beta1, C1_);

    // 6. GEMM2: Y2 = Y1 * W2^T + b2  (M x 256 x 128), Y2 reuses H region
    gemm_wmma_kernel<<<dim3(C2_ / TN_, M_ / TM_), 256, 0, stream>>>(
        Y1, W2, b2, Y2, C1_, C2_);

    // 7. BN2 stats + fused normalize/relu/transpose to [B,128,N]
    colstats_kernel<<<64, C2_, 0, stream>>>(Y2, s2_sum, s2_sq, C2_, M_ / 64);
    finalize_stats_kernel<<<1, C2_, 0, stream>>>(s2_sum, s2_sq, s2_mu, s2_is,
                                                 1.0f / (float)M_);
    bnrelu_transpose_out_kernel<<<dim3(N_ / 32, C2_ / 32, B_), dim3(32, 8), 0,
                                  stream>>>(Y2, s2_mu, s2_is, gamma2, beta2, out);
}